// LBPLayer_47949014892678
// MI455X (gfx1250) — compile-verified
//
#include <hip/hip_runtime.h>
#include <hip/hip_bf16.h>
#include <stdint.h>

// LBP (P=8, R=1, 'uniform') on (32,3,512,512) f32.
// Memory-bound stencil: ~201 MB min traffic -> ~8.6 us @ 23.3 TB/s.
// Strategy: per-block TDM (tensor_load_to_lds) DMA of a 130x18 f32 halo tile
// into LDS, quantize once to u8 in LDS, then branch-free bitmask LBP compute
// with coalesced float4 stores.

typedef __attribute__((ext_vector_type(4))) unsigned int v4u;
typedef __attribute__((ext_vector_type(4))) int          v4i;
typedef __attribute__((ext_vector_type(8))) int          v8i;

#define IMG_H   512
#define IMG_W   512
#define OUT_TW  128          // output tile width per block
#define OUT_TH  16           // output tile height per block
#define TILE_W  (OUT_TW + 2) // 130 (halo)
#define TILE_H  (OUT_TH + 2) // 18  (halo)
#define QPITCH  132          // byte-tile pitch (multiple of 4)

__global__ __launch_bounds__(256) void lbp_tdm_kernel(const float* __restrict__ x,
                                                      float* __restrict__ out) {
    __shared__ float         ftile[TILE_H * TILE_W];   // 9360 B, written by TDM
    __shared__ unsigned char qtile[TILE_H * QPITCH];   // 2376 B, quantized u8

    const int plane = blockIdx.z;                 // b*C + c
    const int x0 = blockIdx.x * OUT_TW;
    const int y0 = blockIdx.y * OUT_TH;

    // Clamp tile origin so the DMA never leaves this 512x512 plane.
    int gsx = x0 - 1; if (gsx < 0) gsx = 0; if (gsx > IMG_W - TILE_W) gsx = IMG_W - TILE_W;
    int gsy = y0 - 1; if (gsy < 0) gsy = 0; if (gsy > IMG_H - TILE_H) gsy = IMG_H - TILE_H;

    const float* plane_base = x + (size_t)plane * (IMG_H * IMG_W);

    // ---- Phase 1: TDM async 2D tile DMA (wave 0 only issues) ----
    if (threadIdx.y == 0) {  // blockDim.x==32 -> exactly wave 0 under wave32
        const uint64_t gaddr = (uint64_t)(uintptr_t)(plane_base + (size_t)gsy * IMG_W + gsx);
        const uint32_t ldsaddr = (uint32_t)(uintptr_t)(&ftile[0]); // low 32b = LDS byte offset

        // D# group 0 (ISA 8.3): count=1 | lds_addr | global_addr[56:0] | type=2
        v4u g0 = { 1u,
                   ldsaddr,
                   (uint32_t)(gaddr & 0xFFFFFFFFu),
                   (uint32_t)((gaddr >> 32) & 0x1FFFFFFu) | (2u << 30) };

        // D# group 1 (ISA 8.4):
        //  wg_mask=0, data_size=2 (4B), no barrier/iterate/pad
        //  tensor_dim0=512, tensor_dim1=512, tile_dim0=130, tile_dim1=18,
        //  tile_dim2=0, tensor_dim0_stride=512, tensor_dim1_stride=0
        v8i g1 = { (int)(2u << 16),                       // data_size = 4 bytes
                   (int)((uint32_t)IMG_W << 16),          // tensor_dim0[15:0]
                   (int)((uint32_t)IMG_H << 16),          // dim0 hi=0 | tensor_dim1[15:0]
                   (int)((uint32_t)TILE_W << 16),         // dim1 hi=0 | tile_dim0
                   (int)TILE_H,                           // tile_dim1 | tile_dim2=0
                   (int)IMG_W,                            // tensor_dim0_stride[31:0]
                   0, 0 };                                // stride hi / dim1_stride

        v4i g2 = { 0, 0, 0, 0 };                          // 2D tensor: dims 2+ unused
        v4i g3 = { 0, 0, 0, 0 };

#if defined(__clang_major__) && (__clang_major__ >= 23)
        v8i gx = { 0, 0, 0, 0, 0, 0, 0, 0 };
        __builtin_amdgcn_tensor_load_to_lds(g0, g1, g2, g3, gx, 0);
#else
        __builtin_amdgcn_tensor_load_to_lds(g0, g1, g2, g3, 0);
#endif
        __builtin_amdgcn_s_wait_tensorcnt(0);
    }
    __syncthreads();

    // ---- Phase 2: quantize f32 tile -> u8 tile (once per element) ----
    const int tid = threadIdx.y * 32 + threadIdx.x;      // 0..255
    for (int i = tid; i < TILE_H * TILE_W; i += 256) {
        const int r = i / TILE_W;
        const int c = i - r * TILE_W;
        int q = (int)floorf(ftile[i] * 255.0f);
        q = (q < 0) ? 0 : ((q > 255) ? 255 : q);
        qtile[r * QPITCH + c] = (unsigned char)q;
    }
    __syncthreads();

    // ---- Phase 3: LBP compute, 2 rows x 4 cols per thread ----
    // Neighbor order matches reference _OFFSETS (circular, clockwise).
    const int ody[8] = { -1, -1, -1,  0,  1, 1,  1,  0 };
    const int odx[8] = { -1,  0,  1,  1,  1, 0, -1, -1 };

    const int cx = x0 + threadIdx.x * 4;                 // 4 consecutive columns
    float* out_plane = out + (size_t)plane * (IMG_H * IMG_W);

    #pragma unroll
    for (int rr = 0; rr < 2; ++rr) {
        const int y  = y0 + threadIdx.y * 2 + rr;
        const int ly = y - gsy;
        float4 res;
        float* resp = &res.x;
        #pragma unroll
        for (int j = 0; j < 4; ++j) {
            const int xx = cx + j;
            const int lx = xx - gsx;
            const int qc = qtile[ly * QPITCH + lx];
            unsigned m = 0u;
            #pragma unroll
            for (int i = 0; i < 8; ++i) {
                const int ny = y + ody[i];
                const int nx = xx + odx[i];
                const int inb = ((unsigned)ny < (unsigned)IMG_H) &
                                ((unsigned)nx < (unsigned)IMG_W);
                const int nv = inb ? (int)qtile[(ly + ody[i]) * QPITCH + (lx + odx[i])] : 0;
                m |= ((nv >= qc) ? 1u : 0u) << i;
            }
            const unsigned rot = ((m << 1) | (m >> 7)) & 0xFFu;
            const int trans = __popc((m ^ rot) & 0xFFu);
            const int ones  = __popc(m);
            const int lbp   = (trans <= 2) ? ones : 9;
            resp[j] = (float)lbp * (1.0f / 255.0f);
        }
        // 16B-aligned coalesced store (cx is a multiple of 4)
        *reinterpret_cast<float4*>(out_plane + (size_t)y * IMG_W + cx) = res;
    }
}

extern "C" void kernel_launch(void* const* d_in, const int* in_sizes, int n_in,
                              void* d_out, int out_size, void* d_ws, size_t ws_size,
                              hipStream_t stream) {
    const float* x = (const float*)d_in[0];
    float* out = (float*)d_out;

    const int planes = in_sizes[0] / (IMG_H * IMG_W);    // B*C = 96
    dim3 grid(IMG_W / OUT_TW, IMG_H / OUT_TH, planes);   // 4 x 32 x 96
    dim3 block(32, 8);                                   // 256 threads = 8 waves (wave32)
    lbp_tdm_kernel<<<grid, block, 0, stream>>>(x, out);
}